// Generator_146028888230
// MI455X (gfx1250) — compile-verified
//
#include <hip/hip_runtime.h>
#include <hip/hip_bf16.h>
#include <math.h>

#define CC 128
#define NN 32768
#define KK 20
#define BLKCH 8                 // blocks per channel
#define SPB (NN / BLKCH)        // 4096 samples per block
#define NBLK2 (CC * BLKCH)      // 1024 blocks in main kernel
#define EPSF 1e-7f

typedef float v2f __attribute__((ext_vector_type(2)));
typedef float v8f __attribute__((ext_vector_type(8)));

#if defined(__HIP_DEVICE_COMPILE__) && __has_builtin(__builtin_amdgcn_wmma_f32_16x16x4_f32)
#define HAVE_WMMA_F32X4 1
#endif

__device__ __forceinline__ float fast_tanh(float x) {
#if defined(__HIP_DEVICE_COMPILE__)
#if __has_builtin(__builtin_amdgcn_tanhf)
  return __builtin_amdgcn_tanhf(x);
#elif __has_builtin(__builtin_amdgcn_tanh_f32)
  return __builtin_amdgcn_tanh_f32(x);
#else
  float e = __builtin_amdgcn_exp2f(x * 2.8853900817779268f);  // 2*log2(e)
  return 1.0f - 2.0f / (e + 1.0f);
#endif
#else
  return tanhf(x);
#endif
}

__device__ __forceinline__ v2f ld_nt_v2f(const float* p) {
#if defined(__HIP_DEVICE_COMPILE__) && __has_builtin(__builtin_nontemporal_load)
  return __builtin_nontemporal_load((const v2f*)p);
#else
  return *(const v2f*)p;
#endif
}

// order-preserving float <-> uint keys for atomic min/max
__device__ __forceinline__ unsigned f2key(float f) {
  unsigned u = __float_as_uint(f);
  return (u & 0x80000000u) ? ~u : (u | 0x80000000u);
}
__device__ __forceinline__ float key2f(unsigned k) {
  unsigned u = (k & 0x80000000u) ? (k ^ 0x80000000u) : ~k;
  return __uint_as_float(u);
}

// ---------------------------------------------------------------- kernel 1
// pts = tanh(tanh(zf@Wf1+bf1)@Wf2+bf2); sort; direction flip -> yp[C][K]
__global__ __launch_bounds__(128) void genf_kernel(
    const float* __restrict__ zf, const float* __restrict__ Wf1,
    const float* __restrict__ bf1, const float* __restrict__ Wf2,
    const float* __restrict__ bf2, const int* __restrict__ dirs,
    float* __restrict__ yp, unsigned* __restrict__ minkey,
    unsigned* __restrict__ maxkey) {
  const int c = threadIdx.x;
  if (c >= CC) return;
  const float* z = zf + c * 8;
  float hbuf[8];
#pragma unroll
  for (int j = 0; j < 8; ++j) {
    float a = bf1[j];
#pragma unroll
    for (int k = 0; k < 8; ++k) a += z[k] * Wf1[k * 8 + j];
    hbuf[j] = fast_tanh(a);
  }
  float p[KK];
  for (int t = 0; t < KK; ++t) {
    float a = bf2[t];
#pragma unroll
    for (int j = 0; j < 8; ++j) a += hbuf[j] * Wf2[j * KK + t];
    p[t] = fast_tanh(a);
  }
  for (int i = 1; i < KK; ++i) {  // insertion sort ascending
    float key = p[i];
    int j = i - 1;
    while (j >= 0 && p[j] > key) { p[j + 1] = p[j]; --j; }
    p[j + 1] = key;
  }
  const bool d = dirs[c] != 0;
  for (int k = 0; k < KK; ++k) yp[c * KK + k] = d ? p[k] : p[KK - 1 - k];
  minkey[c] = 0xFFFFFFFFu;
  maxkey[c] = 0u;
}

// ---------------------------------------------------------------- kernel 2
#if defined(HAVE_WMMA_F32X4)
// All-WMMA 16-sample MLP tile, fully shuffle-free:
//  L1 (transposed): Ht = tanh(W1permT x Zt + b1)  -> 2x v_wmma_f32_16x16x4_f32
//     W1T rows permuted so hidden {0,1,4,5} sit in rows 0-3 (low half) and
//     {2,3,6,7} in rows 8-11 (high half): D vgprs 0..3 are then EXACTLY the
//     B-operand layout (B[k=v+2h][s]) the layer-2 WMMA needs.
//  L2: S = w2_row0 x Ht -> 2x WMMA; S[s] lands in D vgpr0, lanes 0..15.
struct TileConsts {
  v2f a10, a11;  // layer-1 A operand (permuted W1^T)
  v2f a20, a21;  // layer-2 A operand (w2 in row 0)
  v8f cin;       // layer-1 C init (bias b1, permuted rows)
  float b2;      // layer-2 bias
};

__device__ __forceinline__ float mlp_tile(const float* zrow,
                                          const TileConsts& tc) {
  v2f bz0 = ld_nt_v2f(zrow);      // Z^T tile, k = 2h, 2h+1
  v2f bz1 = ld_nt_v2f(zrow + 4);  // Z^T tile, k = 4+2h, 5+2h
  v8f ht = __builtin_amdgcn_wmma_f32_16x16x4_f32(false, tc.a10, false, bz0,
                                                 (short)0, tc.cin, false, false);
  ht = __builtin_amdgcn_wmma_f32_16x16x4_f32(false, tc.a11, false, bz1,
                                             (short)0, ht, false, false);
  v2f t0, t1;  // tanh of the 4 live rows; these ARE the layer-2 B operands
  t0[0] = fast_tanh(ht[0]);
  t0[1] = fast_tanh(ht[1]);
  t1[0] = fast_tanh(ht[2]);
  t1[1] = fast_tanh(ht[3]);
  v8f zero = {0.f, 0.f, 0.f, 0.f, 0.f, 0.f, 0.f, 0.f};
  v8f d = __builtin_amdgcn_wmma_f32_16x16x4_f32(false, tc.a20, false, t0,
                                                (short)0, zero, false, false);
  d = __builtin_amdgcn_wmma_f32_16x16x4_f32(false, tc.a21, false, t1, (short)0,
                                            d, false, false);
  return fast_tanh(d[0] + tc.b2);  // valid in lanes 0..15 (row 0 of D)
}

// build per-lane constants for one network
__device__ __forceinline__ TileConsts make_tc(const float* W1, const float* b1,
                                              const float* W2, float b2, int m,
                                              int h) {
  TileConsts tc;
  // row m of permuted W1^T: rows 0-3 -> hidden {0,1,4,5}; rows 8-11 -> {2,3,6,7}
  const bool lowrow = (m < 4);
  const bool hirow = (m >= 8) && (m < 12);
  const int q = lowrow ? m : (m - 8);
  const int j = (q & 1) + 4 * (q >> 1) + (hirow ? 2 : 0);
  const bool valid = lowrow || hirow;
  tc.a10[0] = valid ? W1[(2 * h + 0) * 8 + j] : 0.f;
  tc.a10[1] = valid ? W1[(2 * h + 1) * 8 + j] : 0.f;
  tc.a11[0] = valid ? W1[(2 * h + 4) * 8 + j] : 0.f;
  tc.a11[1] = valid ? W1[(2 * h + 5) * 8 + j] : 0.f;
  // layer-2 A: w2 in row 0 only
  tc.a20[0] = (m == 0) ? W2[2 * h + 0] : 0.f;
  tc.a20[1] = (m == 0) ? W2[2 * h + 1] : 0.f;
  tc.a21[0] = (m == 0) ? W2[2 * h + 4] : 0.f;
  tc.a21[1] = (m == 0) ? W2[2 * h + 5] : 0.f;
  // layer-1 bias along permuted rows: vgpr v, half hh -> row v+8hh
#pragma unroll
  for (int v = 0; v < 8; ++v) {
    const int jj = (v & 1) + 4 * (v >> 1) + (h ? 2 : 0);  // jA(v) / jB(v)
    tc.cin[v] = (v < 4) ? b1[jj] : 0.f;
  }
  tc.b2 = b2;
  return tc;
}
#endif

__global__ __launch_bounds__(256) void mlp_main_kernel(
    const float* __restrict__ zx, const float* __restrict__ ze,
    const float* __restrict__ Wx1, const float* __restrict__ bx1,
    const float* __restrict__ Wx2, const float* __restrict__ bx2,
    const float* __restrict__ We1, const float* __restrict__ be1,
    const float* __restrict__ We2, const float* __restrict__ be2,
    float* __restrict__ xout, float* __restrict__ eout,
    unsigned* __restrict__ minkey, unsigned* __restrict__ maxkey,
    float* __restrict__ partials) {
  const int tid = threadIdx.x;
  const int c = blockIdx.x / BLKCH;
  const int chunk = blockIdx.x % BLKCH;
  float mn = 3.4e38f, mx = -3.4e38f, s1 = 0.f, s2 = 0.f;

#if defined(HAVE_WMMA_F32X4)
  const int wave = tid >> 5;
  const int lane = tid & 31;
  const int m = lane & 15;
  const int h = lane >> 4;
  const bool low = (lane < 16);
  const TileConsts tcx = make_tc(Wx1, bx1, Wx2, bx2[0], m, h);
  const TileConsts tce = make_tc(We1, be1, We2, be2[0], m, h);

  int n0 = chunk * SPB + wave * (SPB / 8);  // 512-sample run per wave
  const size_t rowbase = (size_t)c * NN * 8;
#pragma unroll 2
  for (int it = 0; it < (SPB / 8) / 16; ++it, n0 += 16) {
    const size_t off = rowbase + (size_t)(n0 + m) * 8 + 2 * h;
    const float ox = mlp_tile(zx + off, tcx);
    const float oe = mlp_tile(ze + off, tce);
    mn = fminf(mn, low ? ox : 3.4e38f);
    mx = fmaxf(mx, low ? ox : -3.4e38f);
    s1 += low ? oe : 0.f;
    s2 += low ? oe * oe : 0.f;
    if (low) {
      const size_t oidx = (size_t)(n0 + m) * CC + c;  // [n][c] layout
      xout[oidx] = ox;
      eout[oidx] = oe;
    }
  }
#else
  // scalar fallback: one sample per thread
  for (int r = 0; r < SPB / 256; ++r) {
    const int n = chunk * SPB + r * 256 + tid;
    const float* zr = zx + ((size_t)c * NN + n) * 8;
    const float* er = ze + ((size_t)c * NN + n) * 8;
    float ax = bx2[0], ae = be2[0];
#pragma unroll
    for (int j = 0; j < 8; ++j) {
      float hx = bx1[j], he = be1[j];
#pragma unroll
      for (int k = 0; k < 8; ++k) {
        hx += zr[k] * Wx1[k * 8 + j];
        he += er[k] * We1[k * 8 + j];
      }
      ax += fast_tanh(hx) * Wx2[j];
      ae += fast_tanh(he) * We2[j];
    }
    float ox = fast_tanh(ax), oe = fast_tanh(ae);
    mn = fminf(mn, ox);
    mx = fmaxf(mx, ox);
    s1 += oe;
    s2 += oe * oe;
    const size_t oidx = (size_t)n * CC + c;
    xout[oidx] = ox;
    eout[oidx] = oe;
  }
#endif

  // deterministic block reductions
  __shared__ float sA[256], sB[256], sC[256], sD[256];
  sA[tid] = mn; sB[tid] = mx; sC[tid] = s1; sD[tid] = s2;
  __syncthreads();
  for (int s = 128; s > 0; s >>= 1) {
    if (tid < s) {
      sA[tid] = fminf(sA[tid], sA[tid + s]);
      sB[tid] = fmaxf(sB[tid], sB[tid + s]);
      sC[tid] += sC[tid + s];
      sD[tid] += sD[tid + s];
    }
    __syncthreads();
  }
  if (tid == 0) {
    atomicMin(&minkey[c], f2key(sA[0]));  // min/max atomics are order-invariant
    atomicMax(&maxkey[c], f2key(sB[0]));
    partials[2 * blockIdx.x + 0] = sC[0];
    partials[2 * blockIdx.x + 1] = sD[0];
  }
}

// ---------------------------------------------------------------- kernel 3
__global__ __launch_bounds__(256) void stats_kernel(
    const float* __restrict__ partials, const unsigned* __restrict__ minkey,
    const unsigned* __restrict__ maxkey, float* __restrict__ chan,
    float* __restrict__ stats) {
  const int tid = threadIdx.x;
  __shared__ double sh1[256], sh2[256];
  double a = 0.0, b = 0.0;
  for (int i = tid; i < NBLK2; i += 256) {  // fixed order -> deterministic
    a += (double)partials[2 * i + 0];
    b += (double)partials[2 * i + 1];
  }
  sh1[tid] = a; sh2[tid] = b;
  __syncthreads();
  for (int s = 128; s > 0; s >>= 1) {
    if (tid < s) { sh1[tid] += sh1[tid + s]; sh2[tid] += sh2[tid + s]; }
    __syncthreads();
  }
  if (tid == 0) {
    const double M = (double)CC * (double)NN;
    const double mean = sh1[0] / M;
    double var = (sh2[0] - sh1[0] * sh1[0] / M) / (M - 1.0);
    if (var < 0.0) var = 0.0;
    stats[0] = (float)mean;
    stats[1] = (float)(0.1 / sqrt(var));  // 0.1 * inv unbiased std
  }
  if (tid < CC) {
    const float mnv = key2f(minkey[tid]);
    const float mxv = key2f(maxkey[tid]);
    chan[2 * tid + 0] = mnv;
    chan[2 * tid + 1] = 1.0f / (mxv - mnv);
  }
}

// ---------------------------------------------------------------- kernel 4
__global__ __launch_bounds__(256) void calibrate_kernel(
    const float* __restrict__ eraw, const float* __restrict__ yp,
    const float* __restrict__ chan, const float* __restrict__ stats,
    float* __restrict__ out) {
  const int idx = blockIdx.x * 256 + threadIdx.x;
  const int c = idx & (CC - 1);
  const float x = out[idx];  // x_raw staged in d_out, overwritten in place
  const float mnv = chan[2 * c + 0];
  const float rcp = chan[2 * c + 1];
  const float xn = (x - mnv) * rcp;
  int i = (int)(xn * (float)(KK - 1));
  i = i < 0 ? 0 : (i > KK - 2 ? KK - 2 : i);
  const float inv = 1.0f / (float)(KK - 1);
  const float x0 = (float)i * inv;
  const float x1 = (float)(i + 1) * inv;
  const float y0 = yp[c * KK + i];
  const float y1 = yp[c * KK + i + 1];
  float y = y0 + (xn - x0) * (y1 - y0) / (x1 - x0 + EPSF);
  y += (eraw[idx] - stats[0]) * stats[1];
  out[idx] = y;
}

// ---------------------------------------------------------------- launch
extern "C" void kernel_launch(void* const* d_in, const int* in_sizes, int n_in,
                              void* d_out, int out_size, void* d_ws,
                              size_t ws_size, hipStream_t stream) {
  const float* zf  = (const float*)d_in[0];
  const float* zx  = (const float*)d_in[1];
  const float* ze  = (const float*)d_in[2];
  const float* Wx1 = (const float*)d_in[3];
  const float* bx1 = (const float*)d_in[4];
  const float* Wx2 = (const float*)d_in[5];
  const float* bx2 = (const float*)d_in[6];
  const float* We1 = (const float*)d_in[7];
  const float* be1 = (const float*)d_in[8];
  const float* We2 = (const float*)d_in[9];
  const float* be2 = (const float*)d_in[10];
  const float* Wf1 = (const float*)d_in[11];
  const float* bf1 = (const float*)d_in[12];
  const float* Wf2 = (const float*)d_in[13];
  const float* bf2 = (const float*)d_in[14];
  const int* dirs  = (const int*)d_in[15];
  float* out = (float*)d_out;

  char* ws = (char*)d_ws;
  float* yp         = (float*)(ws + 0);         // 128*20 f32
  unsigned* minkey  = (unsigned*)(ws + 10240);  // 128 u32
  unsigned* maxkey  = (unsigned*)(ws + 10752);  // 128 u32
  float* chan       = (float*)(ws + 11264);     // 128 * {min, 1/(max-min)}
  float* stats      = (float*)(ws + 12288);     // mean, 0.1/std
  float* partials   = (float*)(ws + 12320);     // NBLK2 * {sum, sumsq}
  float* eraw       = (float*)(ws + 20736);     // NN*CC f32, [n][c]

  genf_kernel<<<1, 128, 0, stream>>>(zf, Wf1, bf1, Wf2, bf2, dirs, yp, minkey,
                                     maxkey);
  mlp_main_kernel<<<NBLK2, 256, 0, stream>>>(zx, ze, Wx1, bx1, Wx2, bx2, We1,
                                             be1, We2, be2, out, eraw, minkey,
                                             maxkey, partials);
  stats_kernel<<<1, 256, 0, stream>>>(partials, minkey, maxkey, chan, stats);
  calibrate_kernel<<<(NN * CC) / 256, 256, 0, stream>>>(eraw, yp, chan, stats,
                                                        out);
}